// CharToVectorLayer5_26233660244455
// MI455X (gfx1250) — compile-verified
//
#include <hip/hip_runtime.h>
#include <hip/hip_bf16.h>
#include <stdint.h>

// ---------------------------------------------------------------------------
// CharToVectorLayer: out[n,d] = sum_{c=0..9} W[c]/55 * table[x[n,c], d]
//   x:     (65536, 10) int32 indices in [0, 8000)
//   table: (8000, 256) f32
//   out:   (65536, 256) f32
//
// CDNA5 TDM gather mode: one tensor_load_to_lds per output row DMAs the ten
// indexed 1KB table rows into LDS (16-bit row indices in D# groups 2/3),
// double-buffered per wave (TENSORcnt pipelining), consumed with
// bank-conflict-free ds_load_b128, weighted-summed in VALU, stored as two
// coalesced global b128 per lane.
// ---------------------------------------------------------------------------

#define N_ROWS_C   10
#define DIM_D      256
#define TABLE_ROWS 8000
#define ROW_F32S   (N_ROWS_C * DIM_D)          // 2560 floats = 10 KB per tile
#define TILE_BYTES (ROW_F32S * 4)

#if defined(__has_builtin)
#  if __has_builtin(__builtin_amdgcn_tensor_load_to_lds) && \
      __has_builtin(__builtin_amdgcn_s_wait_tensorcnt)
#    define HAVE_TDM 1
#  endif
#endif
#ifndef HAVE_TDM
#  define HAVE_TDM 0
#endif

typedef unsigned int v4u __attribute__((ext_vector_type(4)));
typedef int          v4i __attribute__((ext_vector_type(4)));
typedef int          v8i __attribute__((ext_vector_type(8)));

__device__ __constant__ float c_wnorm[10] = {
    10.f / 55.f, 9.f / 55.f, 8.f / 55.f, 7.f / 55.f, 6.f / 55.f,
    5.f / 55.f,  4.f / 55.f, 3.f / 55.f, 2.f / 55.f, 1.f / 55.f};

#if HAVE_TDM
// Issue one TDM gather: 10 rows of `table` (indices from xrow) -> LDS tile.
__device__ __forceinline__ void tdm_gather_rows(const float* table,
                                                const int* xrow,
                                                uint32_t lds_byte_off) {
  // Ten wave-uniform indices, scalarized for the SGPR descriptor.
  uint32_t s[10];
#pragma unroll
  for (int c = 0; c < 10; ++c)
    s[c] = (uint32_t)__builtin_amdgcn_readfirstlane(xrow[c]);

  const uint32_t p0 = (s[0] & 0xFFFFu) | (s[1] << 16);
  const uint32_t p1 = (s[2] & 0xFFFFu) | (s[3] << 16);
  const uint32_t p2 = (s[4] & 0xFFFFu) | (s[5] << 16);
  const uint32_t p3 = (s[6] & 0xFFFFu) | (s[7] << 16);
  const uint32_t p4 = (s[8] & 0xFFFFu) | (s[9] << 16);

  const uint64_t ga = (uint64_t)(uintptr_t)table;

  // D# group 0: count=1 | gather_mode(16-bit idx), lds_addr, 57b global addr, type=2
  v4u g0;
  g0[0] = 0x80000001u;                 // gather_mode=1, idx16, count=1
  g0[1] = lds_byte_off;                // LDS destination (bytes)
  g0[2] = (uint32_t)ga;                // global_addr[31:0]
  g0[3] = ((uint32_t)(ga >> 32) & 0x01FFFFFFu) | 0x80000000u;  // addr[56:32] | type=2

  // D# group 1: data_size=4B, tensor 256x8000, tile 256 x (10 valid indices)
  v8i g1;
  g1[0] = (int)(2u << 16);                   // data_size = 4 bytes
  g1[1] = (int)((uint32_t)DIM_D << 16);      // tensor_dim0[15:0] = 256
  g1[2] = (int)((uint32_t)TABLE_ROWS << 16); // tensor_dim1[15:0] = 8000
  g1[3] = (int)((uint32_t)DIM_D << 16);      // tile_dim0 = 256
  g1[4] = N_ROWS_C;                          // tile_dim1 = #valid gather indices
  g1[5] = DIM_D;                             // tensor_dim0_stride = 256
  g1[6] = 0;
  g1[7] = 0;

  // D# groups 2/3: 16-bit row indices 0..9
  v4i g2; g2[0] = (int)p0; g2[1] = (int)p1; g2[2] = (int)p2; g2[3] = (int)p3;
  v4i g3; g3[0] = (int)p4; g3[1] = 0; g3[2] = 0; g3[3] = 0;

#if __clang_major__ >= 23
  v8i z8 = (v8i)0;
  __builtin_amdgcn_tensor_load_to_lds(g0, g1, g2, g3, z8, 0);
#else
  __builtin_amdgcn_tensor_load_to_lds(g0, g1, g2, g3, 0);
#endif
}

// One wave per output row; 2 waves per block; double-buffered TDM pipeline.
__global__ __launch_bounds__(64) void ctv_tdm_kernel(
    const int* __restrict__ x, const float* __restrict__ table,
    float* __restrict__ out, int nRows, int totalWaves) {
  extern __shared__ float smem[];
  const int lane = threadIdx.x & 31;
  const int wid = threadIdx.x >> 5;                 // 0..1
  const int gw = blockIdx.x * 2 + wid;              // global wave id

  // Two 10KB tiles per wave. Keep all consume-side pointers derived straight
  // from `smem` (addrspace(3)) so the reads lower to ds_load_b128.
  const int base = wid * 2 * ROW_F32S;              // element offset of buf0
  // LDS byte offset of buf0 = low 32 bits of the flat shared address.
  const uint32_t off0 = (uint32_t)(size_t)(const void*)(smem + base);
  const uint32_t off1 = off0 + TILE_BYTES;

  int cur = gw;
  if (cur < nRows) tdm_gather_rows(table, x + (size_t)cur * N_ROWS_C, off0);

  int b = 0;
  while (cur < nRows) {
    const int nxt = cur + totalWaves;
    if (nxt < nRows) {
      tdm_gather_rows(table, x + (size_t)nxt * N_ROWS_C, b ? off0 : off1);
      __builtin_amdgcn_s_wait_tensorcnt(1);   // oldest TDM (our buffer) done
    } else {
      __builtin_amdgcn_s_wait_tensorcnt(0);
    }

    // addrspace(3)-inferable pointer -> ds_load_b128
    const float* sb = smem + base + b * ROW_F32S;
    float4 acc0 = make_float4(0.f, 0.f, 0.f, 0.f);
    float4 acc1 = make_float4(0.f, 0.f, 0.f, 0.f);
#pragma unroll
    for (int c = 0; c < N_ROWS_C; ++c) {
      // lane l covers dwords [4l..4l+3] and [128+4l..+3]: conflict-free b128
      const float4 a0 = *(const float4*)(sb + c * DIM_D + lane * 4);
      const float4 a1 = *(const float4*)(sb + c * DIM_D + 128 + lane * 4);
      const float wc = c_wnorm[c];
      acc0.x = fmaf(wc, a0.x, acc0.x); acc0.y = fmaf(wc, a0.y, acc0.y);
      acc0.z = fmaf(wc, a0.z, acc0.z); acc0.w = fmaf(wc, a0.w, acc0.w);
      acc1.x = fmaf(wc, a1.x, acc1.x); acc1.y = fmaf(wc, a1.y, acc1.y);
      acc1.z = fmaf(wc, a1.z, acc1.z); acc1.w = fmaf(wc, a1.w, acc1.w);
    }
    float4* op = (float4*)(out + (size_t)cur * DIM_D);
    op[lane] = acc0;        // d = 4*lane .. 4*lane+3
    op[lane + 32] = acc1;   // d = 128 + 4*lane ..
    b ^= 1;
    cur = nxt;
  }
}
#endif  // HAVE_TDM

// Fallback: direct global gather (one wave per row, wide b128 loads).
__global__ __launch_bounds__(256) void ctv_direct_kernel(
    const int* __restrict__ x, const float* __restrict__ table,
    float* __restrict__ out, int nRows, int totalWaves) {
  const int lane = threadIdx.x & 31;
  const int gw = (blockIdx.x * blockDim.x + threadIdx.x) >> 5;
  for (int n = gw; n < nRows; n += totalWaves) {
    const int* xr = x + (size_t)n * N_ROWS_C;
    float4 acc0 = make_float4(0.f, 0.f, 0.f, 0.f);
    float4 acc1 = make_float4(0.f, 0.f, 0.f, 0.f);
#pragma unroll
    for (int c = 0; c < N_ROWS_C; ++c) {
      const float* row = table + (size_t)xr[c] * DIM_D;
      const float4 a0 = *(const float4*)(row + lane * 4);
      const float4 a1 = *(const float4*)(row + 128 + lane * 4);
      const float wc = c_wnorm[c];
      acc0.x = fmaf(wc, a0.x, acc0.x); acc0.y = fmaf(wc, a0.y, acc0.y);
      acc0.z = fmaf(wc, a0.z, acc0.z); acc0.w = fmaf(wc, a0.w, acc0.w);
      acc1.x = fmaf(wc, a1.x, acc1.x); acc1.y = fmaf(wc, a1.y, acc1.y);
      acc1.z = fmaf(wc, a1.z, acc1.z); acc1.w = fmaf(wc, a1.w, acc1.w);
    }
    float4* op = (float4*)(out + (size_t)n * DIM_D);
    op[lane] = acc0;
    op[lane + 32] = acc1;
  }
}

extern "C" void kernel_launch(void* const* d_in, const int* in_sizes, int n_in,
                              void* d_out, int out_size, void* d_ws,
                              size_t ws_size, hipStream_t stream) {
  const int* x = (const int*)d_in[0];          // (1024*64, 10) int32
  const float* table = (const float*)d_in[1];  // (8000, 256) f32
  float* out = (float*)d_out;                  // (1024*64, 256) f32
  const int nRows = in_sizes[0] / N_ROWS_C;    // 65536

#if HAVE_TDM
  const int blocks = 2048;                     // 2 waves/block
  const int totalWaves = blocks * 2;
  const size_t shmem = 2 /*waves*/ * 2 /*bufs*/ * (size_t)TILE_BYTES;  // 40 KB
  ctv_tdm_kernel<<<blocks, 64, shmem, stream>>>(x, table, out, nRows,
                                                totalWaves);
#else
  const int blocks = 512;                      // 8 waves/block
  const int totalWaves = blocks * 8;
  ctv_direct_kernel<<<blocks, 256, 0, stream>>>(x, table, out, nRows,
                                                totalWaves);
#endif
}